// WaveLM_4440996184327
// MI455X (gfx1250) — compile-verified
//
#include <hip/hip_runtime.h>
#include <math.h>

// Problem constants (match reference)
#define Bdim 4
#define Tdim 512
#define Vdim 8000
#define Hdim 7
#define BT   (Bdim * Tdim)
#define ROWS 16                      // token rows per pair-kernel block

typedef float v2f __attribute__((ext_vector_type(2)));
typedef float v8f __attribute__((ext_vector_type(8)));

__constant__ __device__ const float kTwoPi = 6.28318530717958647692f;

// ---------------------------------------------------------------------------
// Kernel 1: vocab harmonic tables.
//   vw[h][v]  = 2*pi * f_v * (h+1)
//   vAmp[h][v]= a_v * (h+1)^(-decay)
//   vsin/vcos = sin/cos(vw)
// ---------------------------------------------------------------------------
__global__ void wave_vocab_prep(const float* __restrict__ freq,
                                const float* __restrict__ amp,
                                const float* __restrict__ decayPtr,
                                float* __restrict__ vw,  float* __restrict__ vAmp,
                                float* __restrict__ vsin, float* __restrict__ vcos) {
    int v = blockIdx.x * blockDim.x + threadIdx.x;
    if (v >= Vdim) return;
    float f = freq[v];
    float a = amp[v];
    float d = decayPtr[0];
    #pragma unroll
    for (int h = 1; h <= Hdim; ++h) {
        float w = kTwoPi * f * (float)h;
        float s, c;
        sincosf(w, &s, &c);              // precompute-only: accurate libm version
        float A = a * powf((float)h, -d);
        int o = (h - 1) * Vdim + v;
        vw[o] = w; vAmp[o] = A; vsin[o] = s; vcos[o] = c;
    }
}

// ---------------------------------------------------------------------------
// Kernel 2: per-token tables, gathered from vocab tables.
//   tsA/tcA have the token amplitude pre-folded into sin/cos.
// ---------------------------------------------------------------------------
__global__ void wave_token_prep(const int* __restrict__ ids,
                                const float* __restrict__ vw,  const float* __restrict__ vAmp,
                                const float* __restrict__ vsin, const float* __restrict__ vcos,
                                float* __restrict__ tw,  float* __restrict__ tA,
                                float* __restrict__ tsA, float* __restrict__ tcA) {
    int idx = blockIdx.x * blockDim.x + threadIdx.x;
    if (idx >= BT * Hdim) return;
    int bt = idx / Hdim;
    int h  = idx % Hdim;
    int s  = ids[bt];
    float A = vAmp[h * Vdim + s];
    tw [bt * Hdim + h] = vw[h * Vdim + s];
    tA [bt * Hdim + h] = A;
    tsA[bt * Hdim + h] = A * vsin[h * Vdim + s];
    tcA[bt * Hdim + h] = A * vcos[h * Vdim + s];
}

// ---------------------------------------------------------------------------
// Kernel 3: pairwise interaction C[bt][v] (written straight into d_out).
//   term(i,j) = a_i*A_j * sinc(2*(f_i - f_j))
//             = A_j * ( (a_i sin wi) cos wj - (a_i cos wi) sin wj ) / (wi - wj)
//   with the den==0 limit selecting a_i*A_j.
// One thread per vocab column; ROWS token rows per block. Token tables are
// staged once into LDS (broadcast ds_loads in the hot loop) so the inner
// 7x7 body is pure VALU + one v_rcp_f32 per term.
// ---------------------------------------------------------------------------
__global__ __launch_bounds__(256) void wave_pair_kernel(
        const float* __restrict__ vw,  const float* __restrict__ vAmp,
        const float* __restrict__ vsin, const float* __restrict__ vcos,
        const float* __restrict__ tw,  const float* __restrict__ tA,
        const float* __restrict__ tsA, const float* __restrict__ tcA,
        float* __restrict__ outC) {
    __shared__ float sW[ROWS * Hdim];
    __shared__ float sA[ROWS * Hdim];
    __shared__ float sS[ROWS * Hdim];
    __shared__ float sC[ROWS * Hdim];

    const int tid = threadIdx.x;
    const int r0  = blockIdx.y * ROWS;

    // Stage ROWS*Hdim = 112 floats per table cooperatively (coalesced).
    if (tid < ROWS * Hdim) {
        sW[tid] = tw [r0 * Hdim + tid];
        sA[tid] = tA [r0 * Hdim + tid];
        sS[tid] = tsA[r0 * Hdim + tid];
        sC[tid] = tcA[r0 * Hdim + tid];
    }

    const int v = blockIdx.x * 256 + tid;
    const bool active = (v < Vdim);

    float wv[Hdim], Av[Hdim], sv[Hdim], cv[Hdim];
    #pragma unroll
    for (int j = 0; j < Hdim; ++j) {
        int o = j * Vdim + (active ? v : 0);
        wv[j] = vw[o];
        Av[j] = vAmp[o];
        sv[j] = vsin[o];
        cv[j] = vcos[o];
    }

    __syncthreads();

    for (int rr = 0; rr < ROWS; ++rr) {
        float acc = 0.0f;
        #pragma unroll
        for (int i = 0; i < Hdim; ++i) {
            const float wi = sW[rr * Hdim + i];   // LDS broadcast reads
            const float ai = sA[rr * Hdim + i];
            const float sa = sS[rr * Hdim + i];   // a_i * sin(wi)
            const float ca = sC[rr * Hdim + i];   // a_i * cos(wi)
            #pragma unroll
            for (int j = 0; j < Hdim; ++j) {
                float den = wi - wv[j];
                float num = sa * cv[j] - ca * sv[j];       // a_i*sin(wi-wj)
                float r   = __builtin_amdgcn_rcpf(den);    // v_rcp_f32
                float val = (den == 0.0f) ? ai : num * r;  // sinc limit at 0
                acc = fmaf(Av[j], val, acc);
            }
        }
        if (active) outC[(long long)(r0 + rr) * Vdim + v] = 2.0f * acc;
    }
}

// ---------------------------------------------------------------------------
// Kernel 4: shifted cumsum along T as a strictly-lower-triangular matmul,
// computed with V_WMMA_F32_16X16X4_F32 (exact f32). One wave per
// (batch, 16-wide vocab strip); in-place on d_out.
//
// Layouts (CDNA5 ISA 7.12.2, 32-bit operands):
//   A (16x4):  M = lane&15,  K = vgpr + 2*(lane>>4)
//   B (4x16):  N = lane&15,  K = vgpr + 2*(lane>>4)
//   C/D(16x16):N = lane&15,  M = vgpr + 8*(lane>>4)
// ---------------------------------------------------------------------------
__global__ __launch_bounds__(32) void wave_scan_wmma(float* __restrict__ out) {
    const int lane = threadIdx.x;
    const int half = lane >> 4;     // 0: lanes 0-15, 1: lanes 16-31
    const int n    = lane & 15;     // column / row-of-L index

    const int b = blockIdx.y;
    float* col = out + (long long)b * Tdim * Vdim + blockIdx.x * 16 + n;

    // A operand slices of L_strict: L[m][k] = (k < m), m = lane&15.
    v2f A[4];
    #pragma unroll
    for (int k = 0; k < 4; ++k) {
        A[k].x = ((k * 4 + 0 + 2 * half) < n) ? 1.0f : 0.0f;
        A[k].y = ((k * 4 + 1 + 2 * half) < n) ? 1.0f : 0.0f;
    }

    float carry = 0.0f;   // running column sum of previous tiles (replicated in both halves)

    for (int tile = 0; tile < Tdim / 16; ++tile) {
        const int t0 = tile * 16;

        // Load this wave's B-operand rows of the 16x16 C-tile; lanes 0-15 hold
        // rows {0,1,4,5,8,9,12,13}, lanes 16-31 rows {2,3,6,7,10,11,14,15}.
        v2f Bv[4];
        float partial = 0.0f;
        #pragma unroll
        for (int k = 0; k < 4; ++k) {
            int r0 = t0 + k * 4 + 2 * half;
            float x = col[(long long)r0 * Vdim];
            float y = col[(long long)(r0 + 1) * Vdim];
            Bv[k].x = x; Bv[k].y = y;
            partial += x + y;
        }

        // D = L_strict @ Ctile + carry  (carry broadcast down every row)
        v8f acc;
        #pragma unroll
        for (int r = 0; r < 8; ++r) acc[r] = carry;
        #pragma unroll
        for (int k = 0; k < 4; ++k)
            acc = __builtin_amdgcn_wmma_f32_16x16x4_f32(
                      false, A[k], false, Bv[k], (short)0, acc, false, false);

        // Store logits tile in place (rows of this tile are no longer needed).
        #pragma unroll
        for (int r = 0; r < 8; ++r)
            col[(long long)(t0 + r + 8 * half) * Vdim] = acc[r];

        // carry += full column sum of this C tile (combine the two wave halves)
        float other = __shfl_xor(partial, 16, 32);
        carry += partial + other;
    }
}

// ---------------------------------------------------------------------------
// Launch: vocab prep -> token gather -> pair compute -> WMMA scan (in place)
// Workspace usage: 4*H*V + 4*BT*H floats = ~1.08 MB.
// ---------------------------------------------------------------------------
extern "C" void kernel_launch(void* const* d_in, const int* in_sizes, int n_in,
                              void* d_out, int out_size, void* d_ws, size_t ws_size,
                              hipStream_t stream) {
    (void)in_sizes; (void)n_in; (void)out_size; (void)ws_size;

    const int*   ids   = (const int*)  d_in[0];   // (B,T)
    const float* freq  = (const float*)d_in[1];   // (V,)
    const float* amp   = (const float*)d_in[2];   // (V,)
    const float* decay = (const float*)d_in[3];   // scalar
    // d_in[4] (chunk_size) does not affect the math.

    float* out = (float*)d_out;
    float* ws  = (float*)d_ws;

    const size_t HV = (size_t)Hdim * Vdim;
    float* vw   = ws;
    float* vAmp = ws + HV;
    float* vsin = ws + 2 * HV;
    float* vcos = ws + 3 * HV;
    float* tw   = ws + 4 * HV;
    float* tA   = tw  + (size_t)BT * Hdim;
    float* tsA  = tA  + (size_t)BT * Hdim;
    float* tcA  = tsA + (size_t)BT * Hdim;

    wave_vocab_prep<<<(Vdim + 255) / 256, 256, 0, stream>>>(
        freq, amp, decay, vw, vAmp, vsin, vcos);

    wave_token_prep<<<(BT * Hdim + 255) / 256, 256, 0, stream>>>(
        ids, vw, vAmp, vsin, vcos, tw, tA, tsA, tcA);

    dim3 gridC((Vdim + 255) / 256, BT / ROWS);
    wave_pair_kernel<<<gridC, 256, 0, stream>>>(
        vw, vAmp, vsin, vcos, tw, tA, tsA, tcA, out);

    dim3 gridS(Vdim / 16, Bdim);
    wave_scan_wmma<<<gridS, 32, 0, stream>>>(out);
}